// LocalAttentionCache_32856499815179
// MI455X (gfx1250) — compile-verified
//
#include <hip/hip_runtime.h>
#include <hip/hip_bf16.h>
#include <stdint.h>

// Problem constants from setup_inputs(): positions [4,4096,2] f32, k=32.
#define B_DIM   4
#define L_DIM   4096
#define K_NN    32
#define FEAT    130          // 2*(2*32+1)
#define PI_F    3.14159265358979f

typedef float v2f __attribute__((ext_vector_type(2)));
typedef float v8f __attribute__((ext_vector_type(8)));
typedef unsigned int v4u __attribute__((ext_vector_type(4)));
typedef int  v4i __attribute__((ext_vector_type(4)));
typedef int  v8i __attribute__((ext_vector_type(8)));

// ---------------------------------------------------------------------------
// Kernel 1: brute-force KNN via V_WMMA_F32_16X16X4_F32.
//   dist^2 tile = A(16x4) x B(4x16) with
//     A(cand)  K=0,1 : (-2cx, -2cy)   K=2,3 : (|c|^2, 1)
//     B(query) K=0,1 : ( qx,  qy)     K=2,3 : (1, |q|^2)
//   => D[M=cand][N=query] = |c-q|^2 in a single WMMA, no epilogue.
// Block = 128 thr (4 waves), each wave owns 16 queries; lane pair (l, l+16)
// shares query (l&15), each half scans 8 candidates/tile, keeps a register
// top-32 as packed u64 keys (dist_bits:index), then the pair merges via LDS.
// ---------------------------------------------------------------------------
__global__ __launch_bounds__(128) void knn_topk_kernel(
    const float* __restrict__ positions,
    int*   __restrict__ out_idx,
    float* __restrict__ out_dist)
{
    __shared__ float2   s_pos[L_DIM];     // 32 KB: whole batch's positions
    __shared__ uint64_t s_mk[64 * 64];    // 32 KB: merge scratch (packed keys)

    const int tid  = threadIdx.x;
    const int lane = tid & 31;
    const int wave = tid >> 5;
    const int b    = blockIdx.x >> 6;          // 64 blocks per batch
    const int qblk = (blockIdx.x & 63) * 64;   // first query of this block
    const float* bpos = positions + (size_t)b * L_DIM * 2;

    // ---- Stage batch positions into LDS (TDM if available) ----------------
#if defined(__has_builtin) && __has_builtin(__builtin_amdgcn_tensor_load_to_lds) && __has_builtin(__builtin_amdgcn_s_wait_tensorcnt)
    if (wave == 0) {
        // 1-D tensor descriptor: 8192 dwords (4096 float2) global -> LDS.
        uint64_t ga      = (uint64_t)(uintptr_t)bpos;
        uint32_t lds_off = (uint32_t)(uintptr_t)(&s_pos[0]); // low 32b = LDS byte offset
        v4u g0;
        g0.x = 0x1u;                                   // count=1 user descriptor
        g0.y = lds_off;                                // lds_addr
        g0.z = (uint32_t)(ga & 0xffffffffu);           // global_addr[31:0]
        g0.w = (uint32_t)((ga >> 32) & 0x1ffffffu)     // global_addr[56:32]
             | (2u << 30);                             // type=2 ("image")
        v8i g1;
        g1[0] = (int)(2u << 16);        // data_size=4B; no multicast/pad/barrier
        g1[1] = (int)(8192u << 16);     // tensor_dim0 lo16 at bits 63:48
        g1[2] = (int)(1u << 16);        // tensor_dim0 hi=0, tensor_dim1=1 (lo)
        g1[3] = (int)(8192u << 16);     // tensor_dim1 hi=0, tile_dim0=8192
        g1[4] = 1;                      // tile_dim1=1, tile_dim2=0
        g1[5] = 8192;                   // tensor_dim0_stride lo32
        g1[6] = (int)(8192u << 16);     // stride hi=0, tensor_dim1_stride lo16
        g1[7] = 0;
        v4i z4 = {0, 0, 0, 0};
        v8i z8 = {0, 0, 0, 0, 0, 0, 0, 0};
        __builtin_amdgcn_tensor_load_to_lds(g0, g1, z4, z4, z8, 0);
        __builtin_amdgcn_s_wait_tensorcnt(0);
    }
#else
    for (int i = tid; i < L_DIM; i += 128)
        s_pos[i] = ((const float2*)bpos)[i];
#endif
    __syncthreads();

    const int h  = lane >> 4;                 // half of the lane pair
    const int qr = lane & 15;                 // query row within wave tile
    const int q  = qblk + wave * 16 + qr;     // query index in [0, L)
    const float2 qp = s_pos[q];
    const float  qn = qp.x * qp.x + qp.y * qp.y;

    v2f bfrag;                                // B = query matrix (4x16)
    if (h == 0) { bfrag[0] = qp.x; bfrag[1] = qp.y; }   // K=0,1
    else        { bfrag[0] = 1.0f; bfrag[1] = qn;   }   // K=2,3

    // Ascending top-32 as packed keys: (float_bits(dist) << 32) | cand_index.
    // Positive-float bit patterns order like uints; ties pick the lower index
    // (same tie-break as jax.lax.top_k on -dist).
    uint64_t list[K_NN];
    #pragma unroll
    for (int j = 0; j < K_NN; ++j) list[j] = ~0ull;

    #pragma unroll 2
    for (int cb = 0; cb < L_DIM; cb += 16) {
        const float2 cp = s_pos[cb + qr];     // candidate row this lane feeds
        v2f afrag;                            // A = candidate matrix (16x4)
        if (h == 0) { afrag[0] = -2.0f * cp.x; afrag[1] = -2.0f * cp.y; }
        else        { afrag[0] = cp.x * cp.x + cp.y * cp.y; afrag[1] = 1.0f; }

        v8f acc = {0.f, 0.f, 0.f, 0.f, 0.f, 0.f, 0.f, 0.f};
        acc = __builtin_amdgcn_wmma_f32_16x16x4_f32(
                  false, afrag, false, bfrag, (short)0, acc, false, false);

        #pragma unroll
        for (int r = 0; r < 8; ++r) {
            const int ci = cb + r + h * 8;        // candidate id of acc[r]
            const float d = fmaxf(acc[r], 0.0f);  // clamp WMMA round-off
            uint64_t key = ((uint64_t)__float_as_uint(d) << 32) | (uint32_t)ci;
            if (ci != q && key < list[K_NN - 1]) {
                // Branchless sorted insert: bubble the max forward.
                #pragma unroll
                for (int j = 0; j < K_NN; ++j) {
                    const uint64_t cur = list[j];
                    const bool     lt  = key < cur;
                    list[j] = lt ? key : cur;     // v_cndmask pair
                    key     = lt ? cur : key;     // carry the larger onward
                }
            }
        }
    }

    // ---- Merge the two half-lists of each query through LDS ---------------
    const int mrow = wave * 16 + qr;          // 0..63 within block
    #pragma unroll
    for (int j = 0; j < K_NN; ++j)
        s_mk[mrow * 64 + h * 32 + j] = list[j];
    __syncthreads();

    if (h == 0) {
        const int qg = b * L_DIM + q;
        int*   oi = out_idx  + (size_t)qg * K_NN;
        float* od = out_dist + (size_t)qg * K_NN;
        const uint64_t* ka = &s_mk[mrow * 64];
        int i0 = 0, i1 = 0;
        for (int t = 0; t < K_NN; ++t) {
            uint64_t a = ka[i0], c = ka[32 + i1];
            bool ta = a <= c;
            uint64_t kmin = ta ? a : c;
            i0 += ta ? 1 : 0; i1 += ta ? 0 : 1;
            oi[t] = (int)(uint32_t)kmin;
            od[t] = sqrtf(__uint_as_float((uint32_t)(kmin >> 32)) + 1e-8f);
        }
    }
}

// ---------------------------------------------------------------------------
// Kernel 2: Fourier RPE. One block per query. freqs = linspace(1,32,32)=1..32.
// Feature e: 0 -> t, 1..32 -> sin(t*e*pi), 33..64 -> cos(t*(e-32)*pi),
// repeated for y at +65. self_rpe is the constant row for t=0.
// ---------------------------------------------------------------------------
__global__ __launch_bounds__(256) void rpe_kernel(
    const float* __restrict__ positions,
    const int*   __restrict__ topk,
    float* __restrict__ rpe,
    float* __restrict__ self_rpe)
{
    __shared__ float s_dxc[K_NN];
    __shared__ float s_dyc[K_NN];

    const int qg  = blockIdx.x;               // 0 .. B*L-1
    const int b   = qg >> 12;                 // / 4096
    const int q   = qg & (L_DIM - 1);
    const int tid = threadIdx.x;
    const float2* bpos = (const float2*)positions + (size_t)b * L_DIM;
    const float2  qp   = bpos[q];

    if (tid < K_NN) {
        int    ni = topk[(size_t)qg * K_NN + tid];
        float2 np = bpos[ni];
        float dx = np.x - qp.x, dy = np.y - qp.y;   // NORMALIZE_SCALE = 1
        s_dxc[tid] = dx / (1.0f + fabsf(dx));
        s_dyc[tid] = dy / (1.0f + fabsf(dy));
    }
    __syncthreads();

    float* out = rpe + (size_t)qg * (K_NN * FEAT);
    for (int f = tid; f < K_NN * FEAT; f += 256) {
        unsigned j = (unsigned)f / FEAT;
        int      e = f - (int)j * FEAT;
        float    t = (e < 65) ? s_dxc[j] : s_dyc[j];
        e = (e < 65) ? e : (e - 65);
        float val;
        if      (e == 0)  val = t;
        else if (e <= 32) val = __sinf(t * (float)e * PI_F);
        else              val = __cosf(t * (float)(e - 32) * PI_F);
        out[f] = val;
    }

    float* sout = self_rpe + (size_t)qg * FEAT;
    for (int f = tid; f < FEAT; f += 256) {
        int e = (f < 65) ? f : (f - 65);
        sout[f] = (e > 32) ? 1.0f : 0.0f;     // cos(0)=1 band, everything else 0
    }
}

// ---------------------------------------------------------------------------
extern "C" void kernel_launch(void* const* d_in, const int* in_sizes, int n_in,
                              void* d_out, int out_size, void* d_ws, size_t ws_size,
                              hipStream_t stream)
{
    (void)in_sizes; (void)n_in; (void)out_size; (void)d_ws; (void)ws_size;
    const float* positions = (const float*)d_in[0];   // [4,4096,2] f32
    // d_in[1] is k (=32 per setup_inputs); grid shapes are compile-time here.

    float* out = (float*)d_out;
    const size_t n_idx  = (size_t)B_DIM * L_DIM * K_NN;         // 524288
    const size_t n_rpe  = n_idx * FEAT;                         // 68157440
    const size_t n_self = (size_t)B_DIM * L_DIM * FEAT;         // 2129920

    int*   out_idx  = (int*)out;                 // int32 bits in float slots
    float* out_rpe  = out + n_idx;
    float* out_self = out_rpe + n_rpe;
    float* out_dist = out_self + n_self;

    knn_topk_kernel<<<dim3(B_DIM * (L_DIM / 64)), dim3(128), 0, stream>>>(
        positions, out_idx, out_dist);
    rpe_kernel<<<dim3(B_DIM * L_DIM), dim3(256), 0, stream>>>(
        positions, out_idx, out_rpe, out_self);
}